// FilterDetections_53876069761074
// MI455X (gfx1250) — compile-verified
//
#include <hip/hip_runtime.h>
#include <stdint.h>

// ---------------- problem constants (from reference) ----------------
#define B_IMG   16
#define N_ANCH  49104
#define NCLS    8
#define MAXD    100
#define SCORE_T 0.01f
#define NMS_T   0.5f
#define NEGV    (-1e30f)
#define BLKA    1024          // 32 waves (wave32) per workgroup

typedef __attribute__((ext_vector_type(4))) unsigned int uv4;
typedef __attribute__((ext_vector_type(8))) int          iv8;
typedef __attribute__((ext_vector_type(4))) int          iv4;

// wave32 argmax butterfly, first-index tie-break (matches jnp.argmax)
__device__ __forceinline__ void wave_argmax(float &bv, int &bi) {
#pragma unroll
  for (int off = 16; off > 0; off >>= 1) {
    float ov = __shfl_xor(bv, off, 32);
    int   oi = __shfl_xor(bi, off, 32);
    if (ov > bv || (ov == bv && oi < bi)) { bv = ov; bi = oi; }
  }
}

// =====================================================================
// Kernel A: one workgroup per (image, class). Greedy NMS, 100 rounds.
// Scores for the class column live in LDS (196KB of the WGP's 320KB),
// staged there by the Tensor Data Mover (async DMA, strided 2D tile).
// =====================================================================
__global__ __launch_bounds__(BLKA) void nms_per_class(
    const float* __restrict__ boxes,   // [B,N,4]
    const float* __restrict__ cls,     // [B,N,C]
    float* __restrict__ cand_sc,       // [B*C, MAXD]
    int*   __restrict__ cand_idx) {    // [B*C, MAXD]
  extern __shared__ float smem[];
  float* s_scores = smem;                  // N floats
  float* s_rv     = s_scores + N_ANCH;     // 32
  int*   s_ri     = (int*)(s_rv + 32);     // 32
  float* s_box    = (float*)(s_ri + 32);   // x1,y1,x2,y2,area

  const int b    = blockIdx.x / NCLS;
  const int c    = blockIdx.x % NCLS;
  const int tid  = threadIdx.x;
  const int wave = tid >> 5;

  const float4* boxes4 = (const float4*)(boxes + (size_t)b * N_ANCH * 4);

  // ---- TDM: stage class-score column (stride C=8 floats) into LDS ----
  if (tid == 0) {
    uint64_t gaddr = (uint64_t)(uintptr_t)(cls + ((size_t)b * N_ANCH * NCLS + c));
    uint32_t laddr = (uint32_t)(uintptr_t)s_scores;   // LDS offset (dyn region base)
    uv4 g0; iv8 g1; iv4 g2; iv4 g3; iv8 g4;
    // D# group0: count=1 (valid), is_restore=0, gather=0, lds_addr, global_addr, type=2
    g0.x = 1u;
    g0.y = laddr;
    g0.z = (uint32_t)gaddr;
    g0.w = (uint32_t)((gaddr >> 32) & 0x1FFFFFFull) | (2u << 30);
    // D# group1: wg_mask=0 (not in cluster), data_size=2 (4B), no pad/iter/atomic
    g1 = (iv8)0;
    g1.s0 = (2 << 16);                                   // data_size = 4 bytes
    g1.s1 = (int)(8u << 16);                             // tensor_dim0 = 8 (lo16)
    g1.s2 = (int)(((uint32_t)N_ANCH & 0xFFFFu) << 16);   // dim0 hi=0 | tensor_dim1 lo
    g1.s3 = (int)(((uint32_t)N_ANCH >> 16) | (1u << 16));// tensor_dim1 hi | tile_dim0=1
    g1.s4 = (int)((uint32_t)N_ANCH & 0xFFFFu);           // tile_dim1=N, tile_dim2=0
    g1.s5 = 8;                                           // tensor_dim0_stride = 8 elems
    g1.s6 = 0; g1.s7 = 0;                                // dim1_stride unused (2D tile)
    g2 = (iv4)0; g3 = (iv4)0;                            // dims 3/4 unused
    g4 = (iv8)0;                                         // second descriptor slot: unused
    __builtin_amdgcn_tensor_load_to_lds(g0, g1, g2, g3, g4, 0);
    __builtin_amdgcn_s_wait_tensorcnt(0);                // TENSORcnt == 0
  }
  __syncthreads();

  // ---- pre-mask: valid0 = score > SCORE_T  (others start at NEG) ----
  for (int j = tid; j < N_ANCH; j += BLKA)
    if (s_scores[j] <= SCORE_T) s_scores[j] = NEGV;
  // warm L2 with this image's boxes (re-scanned 100x below)
  for (int l = tid; l < (N_ANCH * 16) / 128; l += BLKA)
    __builtin_prefetch((const char*)boxes4 + (size_t)l * 128, 0, 0);
  __syncthreads();

  float* outsc = cand_sc  + (size_t)blockIdx.x * MAXD;
  int*   outix = cand_idx + (size_t)blockIdx.x * MAXD;

  for (int t = 0; t < MAXD; ++t) {
    // --- block-wide argmax over LDS scores ---
    float bv = -3.4e38f; int bi = 0x7fffffff;
    for (int j = tid; j < N_ANCH; j += BLKA) {
      float v = s_scores[j];
      if (v > bv) { bv = v; bi = j; }      // ascending scan keeps first max
    }
    wave_argmax(bv, bi);
    if ((tid & 31) == 0) { s_rv[wave] = bv; s_ri[wave] = bi; }
    __syncthreads();
    if (wave == 0) {
      float v = s_rv[tid & 31]; int i = s_ri[tid & 31];
      wave_argmax(v, i);
      if (tid == 0) {
        bool ok = v > SCORE_T;
        outsc[t] = ok ? v : NEGV;          // matches where(ok, s, NEG)
        outix[t] = ok ? i : 0;             // matches where(ok, i, 0)
        float4 tb = boxes4[i];
        s_box[0] = tb.x; s_box[1] = tb.y; s_box[2] = tb.z; s_box[3] = tb.w;
        s_box[4] = (tb.z - tb.x) * (tb.w - tb.y);
      }
    }
    __syncthreads();
    // --- suppression: drop everything with IoU > NMS_T (incl. self) ---
    const float tbx = s_box[0], tby = s_box[1], tbz = s_box[2], tbw = s_box[3];
    const float areaA = s_box[4];
    for (int j = tid; j < N_ANCH; j += BLKA) {
      if (s_scores[j] == NEGV) continue;   // valid never resurrects
      float4 bb = boxes4[j];
      float xx1 = fmaxf(tbx, bb.x), yy1 = fmaxf(tby, bb.y);
      float xx2 = fminf(tbz, bb.z), yy2 = fminf(tbw, bb.w);
      float inter = fmaxf(xx2 - xx1, 0.f) * fmaxf(yy2 - yy1, 0.f);
      float areaB = (bb.z - bb.x) * (bb.w - bb.y);
      float iou = inter / (areaA + areaB - inter + 1e-8f);
      if (iou > NMS_T) s_scores[j] = NEGV;
    }
    __syncthreads();
  }
}

// =====================================================================
// Kernel B: one workgroup per image. Global top-100 over 800 candidates
// (iterative argmax-and-remove == top_k order incl. ties), then gather.
// =====================================================================
__global__ __launch_bounds__(256) void topk_gather(
    const float* __restrict__ boxes, const float* __restrict__ rot,
    const float* __restrict__ trans, const float* __restrict__ cand_sc,
    const int* __restrict__ cand_idx, float* __restrict__ d_out) {
  __shared__ float s_sc[NCLS * MAXD];
  __shared__ float s_rv[8];
  __shared__ int   s_ri[8];
  __shared__ int   s_self[MAXD];
  __shared__ float s_sels[MAXD];

  const int b = blockIdx.x, tid = threadIdx.x, wave = tid >> 5;
  const int TOT = NCLS * MAXD;

  for (int j = tid; j < TOT; j += 256) s_sc[j] = cand_sc[(size_t)b * TOT + j];
  __syncthreads();

  for (int t = 0; t < MAXD; ++t) {
    float bv = -3.4e38f; int bi = 0x7fffffff;
    for (int j = tid; j < TOT; j += 256) {
      float v = s_sc[j];
      if (v > bv) { bv = v; bi = j; }
    }
    wave_argmax(bv, bi);
    if ((tid & 31) == 0) { s_rv[wave] = bv; s_ri[wave] = bi; }
    __syncthreads();
    if (wave == 0) {
      int l = tid & 31;
      float v = (l < 8) ? s_rv[l] : -3.4e38f;
      int   i = (l < 8) ? s_ri[l] : 0x7fffffff;
      wave_argmax(v, i);
      if (tid == 0) { s_self[t] = i; s_sels[t] = v; s_sc[i] = NEGV; }
    }
    __syncthreads();
  }

  // d_out = concat(boxes[B,100,4], scores[B,100], labels[B,100], rot[B,100,3], trans[B,100,3])
  float* out_boxes  = d_out;
  float* out_scores = d_out + (size_t)B_IMG * MAXD * 4;
  float* out_labels = out_scores + (size_t)B_IMG * MAXD;
  float* out_rot    = out_labels + (size_t)B_IMG * MAXD;
  float* out_trans  = out_rot + (size_t)B_IMG * MAXD * 3;

  for (int t = tid; t < MAXD; t += 256) {
    int f = s_self[t]; float s = s_sels[t];
    bool valid = s > SCORE_T;
    int label  = f / MAXD;
    int anchor = cand_idx[(size_t)b * TOT + f];
    int safe   = valid ? anchor : 0;
    const float* bx = boxes + ((size_t)b * N_ANCH + safe) * 4;
    const float* rr = rot   + ((size_t)b * N_ANCH + safe) * 3;
    const float* tt = trans + ((size_t)b * N_ANCH + safe) * 3;
    float* ob = out_boxes + ((size_t)b * MAXD + t) * 4;
    ob[0] = valid ? bx[0] : -1.f; ob[1] = valid ? bx[1] : -1.f;
    ob[2] = valid ? bx[2] : -1.f; ob[3] = valid ? bx[3] : -1.f;
    out_scores[b * MAXD + t] = valid ? s : -1.f;
    out_labels[b * MAXD + t] = valid ? (float)label : -1.f;
    float* orr = out_rot + ((size_t)b * MAXD + t) * 3;
    orr[0] = valid ? rr[0] : -1.f; orr[1] = valid ? rr[1] : -1.f; orr[2] = valid ? rr[2] : -1.f;
    float* ot = out_trans + ((size_t)b * MAXD + t) * 3;
    ot[0] = valid ? tt[0] : -1.f; ot[1] = valid ? tt[1] : -1.f; ot[2] = valid ? tt[2] : -1.f;
  }
}

// =====================================================================
extern "C" void kernel_launch(void* const* d_in, const int* in_sizes, int n_in,
                              void* d_out, int out_size, void* d_ws, size_t ws_size,
                              hipStream_t stream) {
  (void)in_sizes; (void)n_in; (void)out_size; (void)ws_size;
  const float* boxes = (const float*)d_in[0];
  const float* cls   = (const float*)d_in[1];
  const float* rot   = (const float*)d_in[2];
  const float* trans = (const float*)d_in[3];

  float* cand_sc  = (float*)d_ws;                                  // 12800 f32
  int*   cand_idx = (int*)(cand_sc + (size_t)B_IMG * NCLS * MAXD); // 12800 i32

  size_t smemA = ((size_t)N_ANCH + 80) * sizeof(float);  // ~196.7KB of 320KB/WGP
  (void)hipFuncSetAttribute((const void*)nms_per_class,
                            hipFuncAttributeMaxDynamicSharedMemorySize, (int)smemA);

  hipLaunchKernelGGL(nms_per_class, dim3(B_IMG * NCLS), dim3(BLKA), smemA, stream,
                     boxes, cls, cand_sc, cand_idx);
  hipLaunchKernelGGL(topk_gather, dim3(B_IMG), dim3(256), 0, stream,
                     boxes, rot, trans, cand_sc, cand_idx, (float*)d_out);
}